// Continuity_loss_67654324847173
// MI455X (gfx1250) — compile-verified
//
#include <hip/hip_runtime.h>
#include <cstdint>
#include <cstddef>

typedef __attribute__((ext_vector_type(2))) float v2f;
typedef __attribute__((ext_vector_type(8))) float v8f;

namespace {
constexpr int ROWS    = 8192;
constexpr int COLS    = 4096;
constexpr int TR      = 6;           // compute rows per block
constexpr int HALO    = 2;           // neighbor - 1
constexpr int SROWS   = TR + HALO;   // staged rows (128 KB LDS tile)
constexpr int THREADS = 256;         // 8 wave32s
constexpr int WAVES   = THREADS / 32;
constexpr int C4      = COLS / 4;    // float4 columns per row
}

// f(v) = v * ln(v) with v = |x*y|   (matches jnp: abs -> c*log(c))
__device__ __forceinline__ float xlogx_term(float x, float y) {
    float v = fabsf(x * y);
    return v * __logf(v);
}

// Wave-level sum of 32 per-lane partials using V_WMMA_F32_16X16X4_F32.
// A-VGPR0 = acc  -> A[M=l, K=0] (l<16) / A[M=l-16, K=2] (l>=16); A-VGPR1 = 0.
// B = all ones (layout-independent) => D[m,n] = acc_m + acc_{m+16}.
// Sum of D's 8 VGPRs per lane = half-sum; shfl_xor(16) combines halves.
// EXEC must be all ones at the call site.
__device__ __forceinline__ float wave_reduce_wmma(float acc) {
    v2f A; A[0] = acc;  A[1] = 0.0f;
    v2f B; B[0] = 1.0f; B[1] = 1.0f;
    v8f C = {0.0f, 0.0f, 0.0f, 0.0f, 0.0f, 0.0f, 0.0f, 0.0f};
    v8f D = __builtin_amdgcn_wmma_f32_16x16x4_f32(
        /*neg_a=*/false, A, /*neg_b=*/false, B,
        /*c_mod=*/(short)0, C, /*reuse_a=*/false, /*reuse_b=*/false);
    float t = 0.0f;
#pragma unroll
    for (int k = 0; k < 8; ++k) t += D[k];
    t += __shfl_xor(t, 16, 32);
    return t;                         // full wave sum in every lane
}

__global__ __launch_bounds__(THREADS)
void cont_loss_partial(const float* __restrict__ X, float* __restrict__ partial)
{
    __shared__ float smem[SROWS * COLS];   // 8 rows x 4096 f32 = 131072 B
    __shared__ float wred[WAVES];

    const int tid      = threadIdx.x;
    const int r0       = blockIdx.x * TR;
    const int rowsLeft = ROWS - r0;
    const int nrows    = rowsLeft < SROWS ? rowsLeft : SROWS;  // staged rows
    const int lrMax    = rowsLeft < TR    ? rowsLeft : TR;     // compute rows

    // ---- Stage tile Global -> LDS via CDNA5 async-to-LDS (ASYNCcnt) ----
    {
        uint32_t lds   = (uint32_t)(uintptr_t)(void*)smem + (uint32_t)tid * 16u;
        uint32_t voff  = (uint32_t)tid * 16u;
        uint64_t gbase = (uint64_t)(uintptr_t)(X + (size_t)r0 * COLS);
        // one chunk = 256 lanes * 16 B = 4 KB; 4 chunks per 16 KB row
        const int nchunks = nrows * ((COLS * 4) / (THREADS * 16));
        for (int k = 0; k < nchunks; ++k) {
            asm volatile("global_load_async_to_lds_b128 %0, %1, %2"
                         :: "v"(lds), "v"(voff), "s"(gbase)
                         : "memory");
            lds  += THREADS * 16u;
            voff += THREADS * 16u;
        }
        asm volatile("s_wait_asynccnt 0x0" ::: "memory");
    }
    __syncthreads();

    // ---- Sum f(|X[r,c] * X[r+i,c+j]|) over i in {1,2}, j in {0,1,2} ----
    float acc = 0.0f;
    const int slots = lrMax * C4;
    for (int s = tid; s < slots; s += THREADS) {
        const int  lr    = s / C4;
        const int  c4    = s % C4;
        const bool inter = (c4 < C4 - 1);   // columns c4*4+4, +5 exist

        const float* rowA = smem + lr * COLS;
        const float4 a    = reinterpret_cast<const float4*>(rowA)[c4];

#pragma unroll
        for (int i = 1; i <= HALO; ++i) {
            if (lr + i >= nrows) break;     // row r0+lr+i outside matrix
            const float* rowB = smem + (lr + i) * COLS;
            const float4 b    = reinterpret_cast<const float4*>(rowB)[c4];
            float b4v = 0.0f, b5v = 0.0f;
            if (inter) {
                const float2 t = *reinterpret_cast<const float2*>(rowB + c4 * 4 + 4);
                b4v = t.x; b5v = t.y;
            }
            // 9 terms with e+j <= 3: always in-bounds in the column dim
            acc += xlogx_term(a.x, b.x);   // (e=0,j=0)
            acc += xlogx_term(a.x, b.y);   // (0,1)
            acc += xlogx_term(a.x, b.z);   // (0,2)
            acc += xlogx_term(a.y, b.y);   // (1,0)
            acc += xlogx_term(a.y, b.z);   // (1,1)
            acc += xlogx_term(a.y, b.w);   // (1,2)
            acc += xlogx_term(a.z, b.z);   // (2,0)
            acc += xlogx_term(a.z, b.w);   // (2,1)
            acc += xlogx_term(a.w, b.w);   // (3,0)
            if (inter) {                   // terms touching columns c+4, c+5
                acc += xlogx_term(a.z, b4v);  // (2,2)
                acc += xlogx_term(a.w, b4v);  // (3,1)
                acc += xlogx_term(a.w, b5v);  // (3,2)
            }
        }
    }

    // ---- Block reduction: WMMA wave-sum + tiny cross-wave combine ----
    const float wsum = wave_reduce_wmma(acc);
    if ((tid & 31) == 0) wred[tid >> 5] = wsum;
    __syncthreads();
    if (tid == 0) {
        float s = 0.0f;
#pragma unroll
        for (int k = 0; k < WAVES; ++k) s += wred[k];
        partial[blockIdx.x] = s;
    }
}

__global__ __launch_bounds__(THREADS)
void cont_loss_final(const float* __restrict__ partial, int n, float* __restrict__ out)
{
    __shared__ float wred[WAVES];
    float a = 0.0f;
    for (int i = threadIdx.x; i < n; i += THREADS) a += partial[i];

    const float wsum = wave_reduce_wmma(a);
    if ((threadIdx.x & 31) == 0) wred[threadIdx.x >> 5] = wsum;
    __syncthreads();
    if (threadIdx.x == 0) {
        float s = 0.0f;
#pragma unroll
        for (int k = 0; k < WAVES; ++k) s += wred[k];
        out[0] = -s / (float)ROWS;
    }
}

extern "C" void kernel_launch(void* const* d_in, const int* in_sizes, int n_in,
                              void* d_out, int out_size, void* d_ws, size_t ws_size,
                              hipStream_t stream)
{
    const float* X   = (const float*)d_in[0];   // (8192, 4096) fp32
    float*       out = (float*)d_out;           // scalar loss
    float*       ws  = (float*)d_ws;            // per-block partials

    const int blocks = (ROWS + TR - 1) / TR;    // 1366
    cont_loss_partial<<<blocks, THREADS, 0, stream>>>(X, ws);
    cont_loss_final<<<1, THREADS, 0, stream>>>(ws, blocks, out);

    (void)in_sizes; (void)n_in; (void)out_size; (void)ws_size;
}